// DifferentiableSquareSensor_75892072120707
// MI455X (gfx1250) — compile-verified
//
#include <hip/hip_runtime.h>
#include <hip/hip_bf16.h>

#define IMG_W 2048
#define IMG_H 2048

// x_pix = (x - X0)/DX = (x + 1) * 1024   (DX = 2/2048)
// weight = exp(-0.5*(dx^2+dy^2)/sigma^2), sigma=0.5 -> exp(-2*(dx^2+dy^2))
// exp(-2 d^2) = exp2(C * d^2), C = -2 * log2(e)
#define EXP2_COEF (-2.8853900817779268f)

__device__ __forceinline__ void atomic_add_img(float* p, float v) {
    // Hardware no-return global_atomic_add_f32 at device scope (L2 RMW).
    unsafeAtomicAdd(p, v);
}

__global__ void __launch_bounds__(256)
sensor_splat_kernel(const float* __restrict__ xs,
                    const float* __restrict__ ys,
                    const float* __restrict__ vs,
                    float* __restrict__ img,
                    int n) {
    int i = blockIdx.x * blockDim.x + threadIdx.x;
    if (i >= n) return;

    float xp = (xs[i] + 1.0f) * 1024.0f;   // pixel coords in [0, 2048)
    float yp = (ys[i] + 1.0f) * 1024.0f;
    float v  = vs[i];

    float fx = floorf(xp);
    float fy = floorf(yp);
    int   xi = (int)fx;
    int   yi = (int)fy;
    float xf = xp - fx;                     // frac in [0,1)
    float yf = yp - fy;

    // Separable gaussian taps: wx[k] = exp(-2*(xf-(k-2))^2), via v_exp_f32
    float wx[5], wy[5];
    float sx = 0.0f, sy = 0.0f;
#pragma unroll
    for (int k = 0; k < 5; ++k) {
        float dx = xf - (float)(k - 2);
        float dy = yf - (float)(k - 2);
        float ex = __builtin_amdgcn_exp2f(EXP2_COEF * dx * dx);
        float ey = __builtin_amdgcn_exp2f(EXP2_COEF * dy * dy);
        wx[k] = ex;  sx += ex;
        wy[k] = ey;  sy += ey;
    }
    // sum over all 25 weights == sx*sy (separable); normalize + fold in value
    float scale = v / (sx * sy);

    int x0 = xi - 2;
    int y0 = yi - 2;

    if ((unsigned)x0 <= (unsigned)(IMG_W - 5) &&
        (unsigned)y0 <= (unsigned)(IMG_H - 5)) {
        // Fast path (~99.6% of points): whole 5x5 footprint is in-bounds.
        // 25 unconditional no-return atomics, pipelined on STOREcnt.
        float* base = img + (size_t)y0 * IMG_W + x0;
#pragma unroll
        for (int j = 0; j < 5; ++j) {
            float wyj = wy[j] * scale;
#pragma unroll
            for (int k = 0; k < 5; ++k) {
                atomic_add_img(base + k, wx[k] * wyj);
            }
            base += IMG_W;
        }
    } else {
        // Border path: out-of-range taps contribute exactly 0 in the
        // reference (valid-mask multiply), so skipping them is exact.
#pragma unroll
        for (int j = 0; j < 5; ++j) {
            int py = y0 + j;
            if ((unsigned)py >= (unsigned)IMG_H) continue;
            float wyj = wy[j] * scale;
            float* row = img + (size_t)py * IMG_W;
#pragma unroll
            for (int k = 0; k < 5; ++k) {
                int px = x0 + k;
                if ((unsigned)px < (unsigned)IMG_W) {
                    atomic_add_img(row + px, wx[k] * wyj);
                }
            }
        }
    }
}

__global__ void __launch_bounds__(256)
sensor_zero_kernel(float4* __restrict__ out, int n4) {
    int i = blockIdx.x * blockDim.x + threadIdx.x;
    if (i < n4) {
        out[i] = float4{0.0f, 0.0f, 0.0f, 0.0f};
    }
}

extern "C" void kernel_launch(void* const* d_in, const int* in_sizes, int n_in,
                              void* d_out, int out_size, void* d_ws, size_t ws_size,
                              hipStream_t stream) {
    const float* xs = (const float*)d_in[0];
    const float* ys = (const float*)d_in[1];
    const float* vs = (const float*)d_in[2];
    float* img = (float*)d_out;
    int n = in_sizes[0];

    int n4 = out_size / 4;  // 2048*2048/4 float4s
    int zblocks = (n4 + 255) / 256;
    sensor_zero_kernel<<<zblocks, 256, 0, stream>>>((float4*)img, n4);

    int sblocks = (n + 255) / 256;
    sensor_splat_kernel<<<sblocks, 256, 0, stream>>>(xs, ys, vs, img, n);
}